// moleculeGCN_48352741818636
// MI455X (gfx1250) — compile-verified
//
#include <hip/hip_runtime.h>
#include <hip/hip_bf16.h>

typedef __attribute__((ext_vector_type(2))) float v2f;
typedef __attribute__((ext_vector_type(8))) float v8f;

#define DCH 128
#define GB  256
#define EPS 1e-5f

// ---------------------------------------------------------------------------
// Degree / normalization
// ---------------------------------------------------------------------------
__global__ void k_init_deg(float* deg, int N) {
    int i = blockIdx.x * blockDim.x + threadIdx.x;
    if (i < N) deg[i] = 1.0f;                       // self loop
}

__global__ void k_count_deg(const int* __restrict__ dst, float* deg, int E) {
    int e = blockIdx.x * blockDim.x + threadIdx.x;
    if (e < E)
        __hip_atomic_fetch_add(&deg[dst[e]], 1.0f,
                               __ATOMIC_RELAXED, __HIP_MEMORY_SCOPE_AGENT);
}

__global__ void k_make_dinv(float* deg_inout, int N) {
    int i = blockIdx.x * blockDim.x + threadIdx.x;
    if (i < N) deg_inout[i] = rsqrtf(deg_inout[i]); // deg -> d^{-1/2} in place
}

// ---------------------------------------------------------------------------
// GEMM  H[N x 128] = X[N x 128] * W[128 x 128]   (f32 WMMA 16x16x4)
// block = 256 threads = 8 waves; block owns a 16-row tile staged in LDS;
// wave w owns output columns 16w .. 16w+15.
// ---------------------------------------------------------------------------
__global__ void __launch_bounds__(256)
k_gemm_wmma(const float* __restrict__ X,
            const float* __restrict__ W,
            float* __restrict__ H, int N) {
    __shared__ float xs[16 * DCH];                  // 8 KB X tile

    const int tid  = threadIdx.x;
    const int rowT = blockIdx.x * 16;
    const bool full = (rowT + 16) <= N;             // uniform across block

    // --- cooperative stage: 16x128 floats = 512 float4; 2 per thread ---
    if (full) {
        const float4* gsrc = (const float4*)(X + (size_t)rowT * DCH);
        float4* ldst = (float4*)xs;
        ldst[tid]       = gsrc[tid];
        ldst[tid + 256] = gsrc[tid + 256];
    } else {
#pragma unroll
        for (int j = 0; j < 2; ++j) {
            int f   = tid + j * 256;                // float4 index in tile
            int r   = f >> 5;                       // row in tile (32 float4/row)
            int c4  = f & 31;
            int gr  = rowT + r; if (gr >= N) gr = N - 1;   // clamp (no fault)
            ((float4*)xs)[f] =
                ((const float4*)(X + (size_t)gr * DCH))[c4];
        }
    }
    __syncthreads();

    const int wave = tid >> 5;                      // 0..7 -> column tile
    const int lane = tid & 31;
    const int m    = lane & 15;
    const int kh   = lane >> 4;                     // K-half select
    const int col0 = wave * 16;
    const int n    = col0 + m;

    const float* xrow = xs + m * DCH;               // this lane's A row
    const float* wcol = W + n;                      // this lane's B column

    v8f c = {};
#pragma unroll
    for (int k = 0; k < DCH; k += 4) {
        const int ka = k + 2 * kh;                  // even -> 8B aligned
        v2f a = *(const v2f*)(xrow + ka);           // ds_load_b64
        v2f b;
        b.x = wcol[(size_t)ka * DCH];               // global_load_b32 pair,
        b.y = wcol[(size_t)(ka + 1) * DCH];         // +512B clause partner
        c = __builtin_amdgcn_wmma_f32_16x16x4_f32(
                false, a, false, b, (short)0, c, false, false);
    }

    // C/D layout: VGPR r = rows (r, r+8); lanes 16-31 hold the +8 half.
    float* hrow = H + (size_t)(rowT + 8 * kh) * DCH + n;
    if (full) {
#pragma unroll
        for (int r = 0; r < 8; ++r) hrow[(size_t)r * DCH] = c[r];
    } else {
#pragma unroll
        for (int r = 0; r < 8; ++r) {
            int rr = rowT + 8 * kh + r;
            if (rr < N) H[(size_t)rr * DCH + n] = c[r];
        }
    }
}

// ---------------------------------------------------------------------------
// AGG = H * dinv^2 + bias      (self-loop contribution + bias, also zero-init)
// ---------------------------------------------------------------------------
__global__ void k_self_bias(const float* __restrict__ H,
                            const float* __restrict__ dinv,
                            const float* __restrict__ bias,
                            float* __restrict__ AGG, int N) {
    int i = blockIdx.x * blockDim.x + threadIdx.x;
    if (i < N * DCH) {
        int node = i >> 7;
        int d    = i & (DCH - 1);
        float di = dinv[node];
        AGG[i] = H[i] * di * di + bias[d];
    }
}

// ---------------------------------------------------------------------------
// Edge scatter: AGG[dst] += H[src] * dinv[src]*dinv[dst]
// one wave per edge; lane handles 4 channels (float4 gather, 4 f32 atomics)
// ---------------------------------------------------------------------------
__global__ void k_scatter(const int* __restrict__ src,
                          const int* __restrict__ dst,
                          const float* __restrict__ dinv,
                          const float* __restrict__ H,
                          float* __restrict__ AGG, int E) {
    long long t = (long long)blockIdx.x * blockDim.x + threadIdx.x;
    int e    = (int)(t >> 5);
    int lane = (int)(t & 31);
    if (e >= E) return;
    int s = src[e];
    int d = dst[e];
    float w = dinv[s] * dinv[d];
    float4 v = ((const float4*)(H + (long long)s * DCH))[lane];
    float* out = AGG + (long long)d * DCH + lane * 4;
    __hip_atomic_fetch_add(out + 0, v.x * w, __ATOMIC_RELAXED, __HIP_MEMORY_SCOPE_AGENT);
    __hip_atomic_fetch_add(out + 1, v.y * w, __ATOMIC_RELAXED, __HIP_MEMORY_SCOPE_AGENT);
    __hip_atomic_fetch_add(out + 2, v.z * w, __ATOMIC_RELAXED, __HIP_MEMORY_SCOPE_AGENT);
    __hip_atomic_fetch_add(out + 3, v.w * w, __ATOMIC_RELAXED, __HIP_MEMORY_SCOPE_AGENT);
}

// ---------------------------------------------------------------------------
// BatchNorm statistics: per-channel sum / sumsq (atomic tree over node chunks)
// ---------------------------------------------------------------------------
__global__ void k_zero_stats(float* sum, float* sumsq) {
    int i = threadIdx.x;
    if (i < DCH) { sum[i] = 0.0f; sumsq[i] = 0.0f; }
}

#define STAT_CHUNK 256
__global__ void k_col_stats(const float* __restrict__ AGG,
                            float* sum, float* sumsq, int N) {
    int d     = threadIdx.x;                    // blockDim.x == 128
    int start = blockIdx.x * STAT_CHUNK;
    int stop  = start + STAT_CHUNK; if (stop > N) stop = N;
    float s = 0.0f, ss = 0.0f;
    for (int i = start; i < stop; ++i) {
        float v = AGG[(long long)i * DCH + d];
        s += v; ss += v * v;
    }
    __hip_atomic_fetch_add(&sum[d],   s,  __ATOMIC_RELAXED, __HIP_MEMORY_SCOPE_AGENT);
    __hip_atomic_fetch_add(&sumsq[d], ss, __ATOMIC_RELAXED, __HIP_MEMORY_SCOPE_AGENT);
}

__global__ void k_bn_act(const float* __restrict__ AGG,
                         const float* __restrict__ sum,
                         const float* __restrict__ sumsq,
                         const float* __restrict__ gamma,
                         const float* __restrict__ beta,
                         float* __restrict__ OUT, int N, int relu) {
    int i = blockIdx.x * blockDim.x + threadIdx.x;
    if (i < N * DCH) {
        int d     = i & (DCH - 1);
        float inv = 1.0f / (float)N;
        float mu  = sum[d] * inv;
        float var = sumsq[d] * inv - mu * mu;
        float y   = (AGG[i] - mu) * rsqrtf(var + EPS) * gamma[d] + beta[d];
        OUT[i] = relu ? fmaxf(y, 0.0f) : y;
    }
}

// ---------------------------------------------------------------------------
// Global mean pool
// ---------------------------------------------------------------------------
__global__ void k_zero_pool(float* out, float* cnt) {
    int i = blockIdx.x * blockDim.x + threadIdx.x;
    if (i < GB * DCH) out[i] = 0.0f;
    if (i < GB) cnt[i] = 0.0f;
}

__global__ void k_pool_add(const float* __restrict__ Hf,
                           const int* __restrict__ batch,
                           float* out, float* cnt, int N) {
    long long t = (long long)blockIdx.x * blockDim.x + threadIdx.x;
    int node = (int)(t >> 5);
    int lane = (int)(t & 31);
    if (node >= N) return;
    int seg = batch[node] - batch[0];
    float4 v = ((const float4*)(Hf + (long long)node * DCH))[lane];
    float* o = out + (long long)seg * DCH + lane * 4;
    __hip_atomic_fetch_add(o + 0, v.x, __ATOMIC_RELAXED, __HIP_MEMORY_SCOPE_AGENT);
    __hip_atomic_fetch_add(o + 1, v.y, __ATOMIC_RELAXED, __HIP_MEMORY_SCOPE_AGENT);
    __hip_atomic_fetch_add(o + 2, v.z, __ATOMIC_RELAXED, __HIP_MEMORY_SCOPE_AGENT);
    __hip_atomic_fetch_add(o + 3, v.w, __ATOMIC_RELAXED, __HIP_MEMORY_SCOPE_AGENT);
    if (lane == 0)
        __hip_atomic_fetch_add(&cnt[seg], 1.0f, __ATOMIC_RELAXED, __HIP_MEMORY_SCOPE_AGENT);
}

__global__ void k_pool_div(float* out, const float* cnt) {
    int i = blockIdx.x * blockDim.x + threadIdx.x;
    if (i < GB * DCH) out[i] = out[i] / fmaxf(cnt[i >> 7], 1.0f);
}

// ---------------------------------------------------------------------------
// Host-side layer driver
// ---------------------------------------------------------------------------
static void run_layer(const float* Xin, const float* Wt, const float* bias,
                      const float* gamma, const float* beta,
                      const int* src, const int* dst, const float* dinv,
                      float* H, float* AGG, float* sum, float* sumsq,
                      float* OUT, int N, int E, int relu, hipStream_t stream) {
    const int ND = N * DCH;
    k_gemm_wmma<<<(N + 15) / 16, 256, 0, stream>>>(Xin, Wt, H, N);
    k_self_bias<<<(ND + 255) / 256, 256, 0, stream>>>(H, dinv, bias, AGG, N);
    {
        long long threads = (long long)E * 32;
        k_scatter<<<(unsigned)((threads + 255) / 256), 256, 0, stream>>>(
            src, dst, dinv, H, AGG, E);
    }
    k_zero_stats<<<1, 256, 0, stream>>>(sum, sumsq);
    k_col_stats<<<(N + STAT_CHUNK - 1) / STAT_CHUNK, DCH, 0, stream>>>(AGG, sum, sumsq, N);
    k_bn_act<<<(ND + 255) / 256, 256, 0, stream>>>(AGG, sum, sumsq, gamma, beta, OUT, N, relu);
}

extern "C" void kernel_launch(void* const* d_in, const int* in_sizes, int n_in,
                              void* d_out, int out_size, void* d_ws, size_t ws_size,
                              hipStream_t stream) {
    const float* x     = (const float*)d_in[0];
    const int*   eidx  = (const int*)  d_in[1];
    const int*   batch = (const int*)  d_in[3];   // edge_attr (d_in[2]) unused
    const float* W1  = (const float*)d_in[4];
    const float* b1  = (const float*)d_in[5];
    const float* g1  = (const float*)d_in[6];
    const float* be1 = (const float*)d_in[7];
    const float* Wm  = (const float*)d_in[8];
    const float* bm  = (const float*)d_in[9];
    const float* gm  = (const float*)d_in[10];
    const float* bem = (const float*)d_in[11];
    const float* W2  = (const float*)d_in[12];
    const float* b2  = (const float*)d_in[13];
    const float* g2  = (const float*)d_in[14];
    const float* be2 = (const float*)d_in[15];

    const int N = in_sizes[0] / DCH;
    const int E = in_sizes[1] / 2;
    const int*   src = eidx;       // edge_index[0]
    const int*   dst = eidx + E;   // edge_index[1]
    const long long ND = (long long)N * DCH;

    // workspace layout (floats)
    float* ws    = (float*)d_ws;
    float* dinv  = ws;                  // N   (deg, then d^{-1/2} in place)
    float* H     = dinv  + N;           // N*D
    float* AGG   = H     + ND;          // N*D
    float* B0    = AGG   + ND;          // N*D
    float* B1    = B0    + ND;          // N*D
    float* sum   = B1    + ND;          // D
    float* sumsq = sum   + DCH;         // D
    float* cnt   = sumsq + DCH;         // G

    // degree -> d^{-1/2}
    k_init_deg <<<(N + 255) / 256, 256, 0, stream>>>(dinv, N);
    k_count_deg<<<(E + 255) / 256, 256, 0, stream>>>(dst, dinv, E);
    k_make_dinv<<<(N + 255) / 256, 256, 0, stream>>>(dinv, N);

    // 3 GCN layers (relu on first two)
    run_layer(x,  W1, b1, g1, be1, src, dst, dinv, H, AGG, sum, sumsq, B0, N, E, 1, stream);
    run_layer(B0, Wm, bm, gm, bem, src, dst, dinv, H, AGG, sum, sumsq, B1, N, E, 1, stream);
    run_layer(B1, W2, b2, g2, be2, src, dst, dinv, H, AGG, sum, sumsq, B0, N, E, 0, stream);

    // global mean pool into d_out [G x D]
    float* out = (float*)d_out;
    k_zero_pool<<<(GB * DCH + 255) / 256, 256, 0, stream>>>(out, cnt);
    {
        long long threads = (long long)N * 32;
        k_pool_add<<<(unsigned)((threads + 255) / 256), 256, 0, stream>>>(B0, batch, out, cnt, N);
    }
    k_pool_div<<<(GB * DCH + 255) / 256, 256, 0, stream>>>(out, cnt);
}